// CachedMultiheadAttention_76235669504647
// MI455X (gfx1250) — compile-verified
//
#include <hip/hip_runtime.h>

typedef float v2f __attribute__((ext_vector_type(2)));
typedef float v4f __attribute__((ext_vector_type(4)));
typedef float v8f __attribute__((ext_vector_type(8)));

#define B_     16
#define T_     8
#define E_     2048
#define H_     16
#define HD_    128
#define SC_    4096          // cached sequence length
#define ST_    4104          // total sequence length
#define CHUNK  256
#define PROW   132           // padded partial-row stride (128 acc + m + l + pad)

__device__ __forceinline__ v4f nt_load4(const v4f* p) {
    return __builtin_nontemporal_load(p);
}
__device__ __forceinline__ void nt_store4(v4f* p, v4f v) {
    __builtin_nontemporal_store(v, p);
}

// ---------------------------------------------------------------------------
// Kernel 1: qkv = x @ w_qkv + b_qkv  (M=128, K=2048, N=6144) via WMMA f32
// q -> ws (B,H,T,hd);  k_new/v_new -> tail of output caches (s = 4096..4103)
// ---------------------------------------------------------------------------
__global__ __launch_bounds__(256) void qkv_proj_kernel(
    const float* __restrict__ x, const float* __restrict__ w,
    const float* __restrict__ bias,
    float* __restrict__ qws, float* __restrict__ outk, float* __restrict__ outv)
{
    const int lane = threadIdx.x & 31;
    const int wave = threadIdx.x >> 5;          // M tile (0..7)
    const int lm   = lane & 15;
    const int kk   = (lane >> 4) << 1;          // lanes 0-15 -> K{0,1}; 16-31 -> K{2,3}
    const int m0   = wave * 16;
    const int n0   = blockIdx.x * 16;           // N tile (0..383)

    const float* arow = x + (size_t)(m0 + lm) * E_ + kk;
    const float* bcol = w + (size_t)kk * (3 * E_) + (n0 + lm);

    v8f c = {0.f, 0.f, 0.f, 0.f, 0.f, 0.f, 0.f, 0.f};
    for (int k0 = 0; k0 < E_; k0 += 4) {
        v2f a, b;
        a.x = arow[k0];
        a.y = arow[k0 + 1];
        b.x = bcol[(size_t)k0 * (3 * E_)];
        b.y = bcol[(size_t)(k0 + 1) * (3 * E_)];
        c = __builtin_amdgcn_wmma_f32_16x16x4_f32(false, a, false, b,
                                                  (short)0, c, false, false);
    }

    const float bv  = bias[n0 + lm];
    const int half8 = (lane >> 4) * 8;
    const int n     = n0 + lm;
    const int i3    = n >> 11;                  // 0=q, 1=k, 2=v
    const int rem   = n & 2047;
    const int h     = rem >> 7;
    const int d     = rem & 127;

#pragma unroll
    for (int r = 0; r < 8; ++r) {
        const int row = m0 + half8 + r;         // row = b*T + t
        const int b   = row >> 3;
        const int t   = row & 7;
        const float val = c[r] + bv;
        if (i3 == 0) {
            qws[(size_t)(((b * H_ + h) * T_) + t) * HD_ + d] = val;
        } else {
            float* dst = (i3 == 1) ? outk : outv;
            dst[(size_t)((b * H_ + h) * ST_ + SC_ + t) * HD_ + d] = val;
        }
    }
}

// ---------------------------------------------------------------------------
// Kernel 2: fused cache-copy + flash attention, split over the key dimension
// (flash-decoding). One block per (bh, part). Each k/v row is streamed from
// HBM exactly once (non-temporal), copied to the output cache (non-temporal),
// and consumed for scores / weighted sum in the same pass (online softmax).
// Partial (unnormalized acc, m, l) per (bh, part, t) goes to workspace.
// Thread layout for accumulation: t = tid>>5, dims d = 4*lane .. 4*lane+3.
// ---------------------------------------------------------------------------
__global__ __launch_bounds__(256) void attn_partial_kernel(
    const float* __restrict__ ck, const float* __restrict__ cv,
    const float* __restrict__ qws,
    float* __restrict__ outk, float* __restrict__ outv,
    float* __restrict__ pws, int split)
{
    __shared__ __align__(16) float qs[T_ * HD_];          // 4 KB
    __shared__ float sbuf[T_][CHUNK + 4];                 // ~8 KB

    const int tid  = threadIdx.x;
    const int lane = tid & 31;
    const int wave = tid >> 5;                            // == query index t
    const int bh   = blockIdx.x / split;
    const int part = blockIdx.x - bh * split;
    const int plen = (ST_ + split - 1) / split;
    const int s_begin = part * plen;
    const int s_end   = (s_begin + plen < ST_) ? (s_begin + plen) : ST_;
    const int d0   = lane * 4;
    const float scale = 0.08838834764831845f;             // 1/sqrt(128)

    // stage q (8 x 128 f32, contiguous in ws) into LDS
    {
        const v4f* src = (const v4f*)(qws + (size_t)bh * (T_ * HD_));
        ((v4f*)qs)[tid] = src[tid];
    }
    __syncthreads();

    float m_run = -INFINITY, l_run = 0.f;
    float acc0 = 0.f, acc1 = 0.f, acc2 = 0.f, acc3 = 0.f;

    for (int c0 = s_begin; c0 < s_end; c0 += CHUNK) {
        // ---- Phase A: scores for key s = c0 + tid (and k-row copy) ----
        {
            const int s = c0 + tid;
            float sc[8];
#pragma unroll
            for (int t = 0; t < 8; ++t) sc[t] = -INFINITY;
            if (s < s_end) {
#pragma unroll
                for (int t = 0; t < 8; ++t) sc[t] = 0.f;
                if (s < SC_) {
                    const v4f* krow =
                        (const v4f*)(ck + ((size_t)bh * SC_ + s) * HD_);
                    v4f* kdst =
                        (v4f*)(outk + ((size_t)bh * ST_ + s) * HD_);
                    // prefetch next chunk's k row (4 x 128B lines) into GL2
                    if (s + CHUNK < s_end && s + CHUNK < SC_) {
                        const char* pf =
                            (const char*)(krow) + (size_t)CHUNK * HD_ * 4;
                        __builtin_prefetch(pf, 0, 1);
                        __builtin_prefetch(pf + 128, 0, 1);
                        __builtin_prefetch(pf + 256, 0, 1);
                        __builtin_prefetch(pf + 384, 0, 1);
                    }
                    for (int d4 = 0; d4 < HD_ / 4; ++d4) {
                        const v4f k4 = nt_load4(krow + d4);
                        nt_store4(kdst + d4, k4);
#pragma unroll
                        for (int t = 0; t < 8; ++t) {
                            const v4f q4 = ((const v4f*)qs)[t * 32 + d4];
                            sc[t] += q4.x * k4.x + q4.y * k4.y +
                                     q4.z * k4.z + q4.w * k4.w;
                        }
                    }
                } else {
                    // new-k tail rows, written by kernel 1 (hot in cache)
                    const v4f* krow =
                        (const v4f*)(outk + ((size_t)bh * ST_ + s) * HD_);
                    for (int d4 = 0; d4 < HD_ / 4; ++d4) {
                        const v4f k4 = krow[d4];
#pragma unroll
                        for (int t = 0; t < 8; ++t) {
                            const v4f q4 = ((const v4f*)qs)[t * 32 + d4];
                            sc[t] += q4.x * k4.x + q4.y * k4.y +
                                     q4.z * k4.z + q4.w * k4.w;
                        }
                    }
                }
#pragma unroll
                for (int t = 0; t < 8; ++t) {
                    sc[t] *= scale;
                    if (s - SC_ > t) sc[t] = -INFINITY;   // causal mask
                }
            }
#pragma unroll
            for (int t = 0; t < 8; ++t) sbuf[t][tid] = sc[t];
        }
        __syncthreads();

        // ---- Phase B: per-t online softmax update (wave `wave` owns t) ----
        {
            float sc[8];
            float cmax = -INFINITY;
#pragma unroll
            for (int i = 0; i < 8; ++i) {
                sc[i] = sbuf[wave][lane + i * 32];
                cmax = fmaxf(cmax, sc[i]);
            }
#pragma unroll
            for (int off = 16; off >= 1; off >>= 1)
                cmax = fmaxf(cmax, __shfl_xor(cmax, off, 32));
            const float nm    = fmaxf(m_run, cmax);
            const float alpha = __expf(m_run - nm);
            float lsum = 0.f;
#pragma unroll
            for (int i = 0; i < 8; ++i) {
                const float p = __expf(sc[i] - nm);
                sbuf[wave][lane + i * 32] = p;
                lsum += p;
            }
#pragma unroll
            for (int off = 16; off >= 1; off >>= 1)
                lsum += __shfl_xor(lsum, off, 32);
            l_run = l_run * alpha + lsum;
            m_run = nm;
            acc0 *= alpha; acc1 *= alpha; acc2 *= alpha; acc3 *= alpha;
        }

        // ---- Phase C: acc += p * v (wave 0 also copies v rows) ----
        {
            // prefetch next chunk's v rows into GL2
            {
                const int s2 = c0 + CHUNK + tid;
                if (s2 < s_end && s2 < SC_) {
                    const char* pf =
                        (const char*)(cv + ((size_t)bh * SC_ + s2) * HD_);
                    __builtin_prefetch(pf, 0, 1);
                    __builtin_prefetch(pf + 128, 0, 1);
                    __builtin_prefetch(pf + 256, 0, 1);
                    __builtin_prefetch(pf + 384, 0, 1);
                }
            }
            const int smax = (s_end - c0 < CHUNK) ? (s_end - c0) : CHUNK;
            for (int sl = 0; sl < smax; ++sl) {
                const int s = c0 + sl;
                const float p = sbuf[wave][sl];
                v4f v4;
                if (s < SC_) {
                    v4 = nt_load4(
                        (const v4f*)(cv + ((size_t)bh * SC_ + s) * HD_) + lane);
                    if (wave == 0)
                        nt_store4(
                            (v4f*)(outv + ((size_t)bh * ST_ + s) * HD_) + lane,
                            v4);
                } else {
                    v4 = ((const v4f*)(outv + ((size_t)bh * ST_ + s) * HD_))[lane];
                }
                acc0 += p * v4.x; acc1 += p * v4.y;
                acc2 += p * v4.z; acc3 += p * v4.w;
            }
        }
        __syncthreads();
    }

    // epilogue: unnormalized partial (acc, m, l) for the combine pass
    float* base = pws + ((size_t)blockIdx.x * T_ + wave) * PROW;
    v4f r;
    r.x = acc0; r.y = acc1; r.z = acc2; r.w = acc3;
    *(v4f*)(base + d0) = r;
    if (lane == 0) { base[128] = m_run; base[129] = l_run; }
}

// ---------------------------------------------------------------------------
// Kernel 2b: merge split partials, normalize, write attention output (B,T,E)
// ---------------------------------------------------------------------------
__global__ __launch_bounds__(256) void attn_combine_kernel(
    const float* __restrict__ pws, float* __restrict__ attn_ws, int split)
{
    const int tid  = threadIdx.x;
    const int lane = tid & 31;
    const int wave = tid >> 5;                  // t
    const int bh   = blockIdx.x;
    const int b    = bh >> 4;
    const int h    = bh & 15;
    const int d0   = lane * 4;

    float m_g = -INFINITY;
    for (int p = 0; p < split; ++p)
        m_g = fmaxf(m_g,
                    pws[((size_t)(bh * split + p) * T_ + wave) * PROW + 128]);

    float l_g = 0.f;
    float a0 = 0.f, a1 = 0.f, a2 = 0.f, a3 = 0.f;
    for (int p = 0; p < split; ++p) {
        const float* base =
            pws + ((size_t)(bh * split + p) * T_ + wave) * PROW;
        const float w = __expf(base[128] - m_g);
        const v4f ac = *(const v4f*)(base + d0);
        a0 += w * ac.x; a1 += w * ac.y; a2 += w * ac.z; a3 += w * ac.w;
        l_g += w * base[129];
    }

    const float inv_l = 1.0f / l_g;
    v4f r;
    r.x = a0 * inv_l; r.y = a1 * inv_l; r.z = a2 * inv_l; r.w = a3 * inv_l;
    *(v4f*)(attn_ws + ((size_t)(b * T_ + wave) * E_) + h * HD_ + d0) = r;
}

// ---------------------------------------------------------------------------
// Kernel 3: out = attn @ w_out + b_out  (M=128, K=2048, N=2048) via WMMA f32
// ---------------------------------------------------------------------------
__global__ __launch_bounds__(256) void out_proj_kernel(
    const float* __restrict__ A, const float* __restrict__ w,
    const float* __restrict__ bias, float* __restrict__ out)
{
    const int lane = threadIdx.x & 31;
    const int wave = threadIdx.x >> 5;
    const int lm   = lane & 15;
    const int kk   = (lane >> 4) << 1;
    const int m0   = wave * 16;
    const int n0   = blockIdx.x * 16;           // 0..127

    const float* arow = A + (size_t)(m0 + lm) * E_ + kk;
    const float* bcol = w + (size_t)kk * E_ + (n0 + lm);

    v8f c = {0.f, 0.f, 0.f, 0.f, 0.f, 0.f, 0.f, 0.f};
    for (int k0 = 0; k0 < E_; k0 += 4) {
        v2f a, b;
        a.x = arow[k0];
        a.y = arow[k0 + 1];
        b.x = bcol[(size_t)k0 * E_];
        b.y = bcol[(size_t)(k0 + 1) * E_];
        c = __builtin_amdgcn_wmma_f32_16x16x4_f32(false, a, false, b,
                                                  (short)0, c, false, false);
    }

    const float bv  = bias[n0 + lm];
    const int half8 = (lane >> 4) * 8;
#pragma unroll
    for (int r = 0; r < 8; ++r) {
        const int row = m0 + half8 + r;
        out[(size_t)row * E_ + n0 + lm] = c[r] + bv;
    }
}

// ---------------------------------------------------------------------------
extern "C" void kernel_launch(void* const* d_in, const int* in_sizes, int n_in,
                              void* d_out, int out_size, void* d_ws, size_t ws_size,
                              hipStream_t stream) {
    const float* x     = (const float*)d_in[0];
    const float* ck    = (const float*)d_in[1];
    const float* cv    = (const float*)d_in[2];
    const float* w_qkv = (const float*)d_in[3];
    const float* b_qkv = (const float*)d_in[4];
    const float* w_out = (const float*)d_in[5];
    const float* b_out = (const float*)d_in[6];

    float* out  = (float*)d_out;                              // (B,T,E)
    float* outk = out + (size_t)B_ * T_ * E_;                 // (B,H,ST,hd)
    float* outv = outk + (size_t)B_ * H_ * ST_ * HD_;         // (B,H,ST,hd)

    float* qws     = (float*)d_ws;                            // (B,H,T,hd) 1 MB
    float* attn_ws = qws + (size_t)B_ * H_ * T_ * HD_;        // (B,T,E)    1 MB
    float* pws     = attn_ws + (size_t)B_ * T_ * E_;          // partials

    const size_t fixed = (size_t)B_ * H_ * T_ * HD_ + (size_t)B_ * T_ * E_;
    const size_t need4 = (fixed + (size_t)B_ * H_ * 4 * T_ * PROW) * sizeof(float);
    const int split = (ws_size >= need4) ? 4 : 1;

    qkv_proj_kernel<<<3 * E_ / 16, 256, 0, stream>>>(x, w_qkv, b_qkv,
                                                     qws, outk, outv);
    attn_partial_kernel<<<B_ * H_ * split, 256, 0, stream>>>(
        ck, cv, qws, outk, outv, pws, split);
    attn_combine_kernel<<<B_ * H_, 256, 0, stream>>>(pws, attn_ws, split);
    out_proj_kernel<<<E_ / 16, 256, 0, stream>>>(attn_ws, w_out, b_out, out);
}